// CategoricalEntropyRegLoss_76527727280308
// MI455X (gfx1250) — compile-verified
//
#include <hip/hip_runtime.h>
#include <math.h>

// ---------------- problem constants (match reference) ----------------
#define B      4096
#define FD     256
#define C      32
#define DDIM   2
#define TC     (DDIM * C)     // 64
#define NSEG   (C * C)        // 1024
#define LAMB   0.1f

// ---------------- workspace layout (float offsets) ----------------
#define OFF_FM    0                       // masked normalized features [B][FD]
#define OFF_P     (OFF_FM + B * FD)       // probs [B][TC]
#define OFF_L     (OFF_P + B * TC)        // log-probs [B][TC]
#define OFF_SEGS  (OFF_L + B * TC)        // per-code feature sums [NSEG][FD]
#define OFF_CNT   (OFF_SEGS + NSEG * FD)  // per-code counts [NSEG]
#define OFF_AM    (OFF_CNT + NSEG)        // A = Fm^T P  [FD][TC]
#define OFF_BM    (OFF_AM + FD * TC)      // B = Fm^T L  [FD][TC]
#define OFF_ACC   (OFF_BM + FD * TC)      // scalars + moment vectors (1024 floats)
#define WS_END    (OFF_ACC + 1024)
// ACC sub-offsets
#define A_SM   0   // sum m
#define A_SA   1   // sum m*a
#define A_SE   2   // sum m*e
#define A_SAE  3   // sum m*a*e
#define A_GC   4   // sum A o B
#define A_T2   5   // sum ||s_g||^2 / mx
#define A_T3   6   // sum cnt*||s_g||^2 / mx^2
#define A_VF   16          // [FD]  sum m*fn
#define A_VEF  (16 + FD)   // [FD]  sum m*e*fn
#define A_VP   (16 + 2*FD)           // [TC] sum m*p
#define A_VL   (16 + 2*FD + TC)      // [TC] sum m*l
#define A_VAP  (16 + 2*FD + 2*TC)    // [TC] sum m*a*p
#define A_VAL  (16 + 2*FD + 3*TC)    // [TC] sum m*a*l

typedef __attribute__((ext_vector_type(2))) float v2f;
typedef __attribute__((ext_vector_type(8))) float v8f;

// ---------------------------------------------------------------
// Kernel 1: per-row preprocessing + all rank-1 moment accumulators
// one block of 256 threads per sample row (8 wave32 per block)
// ---------------------------------------------------------------
__global__ __launch_bounds__(FD) void k_prep(
    const float* __restrict__ feat, const float* __restrict__ targ,
    const unsigned char* __restrict__ mask,
    float* __restrict__ fm, float* __restrict__ pP, float* __restrict__ pL,
    float* __restrict__ segs, float* __restrict__ segcnt,
    float* __restrict__ acc)
{
  __shared__ float sh[FD];
  __shared__ float shv[TC];
  __shared__ float shsum[DDIM], shent[DDIM];
  __shared__ int   sharg[DDIM];
  __shared__ float sh_m, sh_e;
  __shared__ int   sh_code;

  const int row = blockIdx.x;
  const int tid = threadIdx.x;

  // ----- feature norm -----
  float f = feat[row * FD + tid];
  sh[tid] = f * f;
  __syncthreads();
  for (int s = FD / 2; s > 0; s >>= 1) {
    if (tid < s) sh[tid] += sh[tid + s];
    __syncthreads();
  }
  const float ssq = sh[0];
  const float cn  = fmaxf(sqrtf(ssq), 1e-12f);
  const float fn  = f / cn;
  const float a   = ssq / (cn * cn);   // == sum(fn*fn)

  if (tid == 0) sh_m = mask[row] ? 1.0f : 0.0f;
  if (tid < TC) shv[tid] = targ[row * TC + tid] + 1e-10f;
  __syncthreads();

  const float mval = sh_m;
  const float fmv  = fn * mval;
  fm[row * FD + tid] = fmv;

  // ----- per-dim normalization + argmax -----
  if (tid < DDIM) {
    float s = 0.0f;
    int am = 0; float mv = shv[tid * C];
    for (int c = 0; c < C; ++c) {
      float v = shv[tid * C + c];
      s += v;
      if (v > mv) { mv = v; am = c; }
    }
    shsum[tid] = s;
    sharg[tid] = am;
  }
  __syncthreads();

  float pv = 0.0f, lv = 0.0f;
  if (tid < TC) {
    const int d = tid >> 5;      // tid / C
    pv = shv[tid] / shsum[d];
    lv = logf(pv);
    pP[row * TC + tid] = pv;
    pL[row * TC + tid] = lv;
    sh[tid] = pv * lv;           // entropy partials (sh reuse after sync)
  }
  __syncthreads();
  if (tid < DDIM) {
    float s = 0.0f;
    for (int c = 0; c < C; ++c) s += sh[tid * C + c];
    shent[tid] = s;
  }
  __syncthreads();
  if (tid == 0) {
    sh_e    = (shent[0] + shent[1]) * (1.0f / DDIM);   // e_j = mean_d ent
    sh_code = sharg[0] + C * sharg[1];                 // pw = [1, 32]
  }
  __syncthreads();
  const float e   = sh_e;
  const int  code = sh_code;

  // ----- accumulators -----
  atomicAdd(&segs[code * FD + tid], fmv);
  atomicAdd(&acc[A_VF  + tid], fmv);
  atomicAdd(&acc[A_VEF + tid], e * fmv);
  if (tid < TC) {
    atomicAdd(&acc[A_VP  + tid], mval * pv);
    atomicAdd(&acc[A_VL  + tid], mval * lv);
    atomicAdd(&acc[A_VAP + tid], mval * a * pv);
    atomicAdd(&acc[A_VAL + tid], mval * a * lv);
  }
  if (tid == 0) {
    atomicAdd(&acc[A_SM],  mval);
    atomicAdd(&acc[A_SA],  mval * a);
    atomicAdd(&acc[A_SE],  mval * e);
    atomicAdd(&acc[A_SAE], mval * a * e);
    atomicAdd(&segcnt[code], mval);
  }
}

// ---------------------------------------------------------------
// Kernel 2: split-K WMMA GEMMs  A = Fm^T P, B = Fm^T L  (fp32 WMMA)
// grid (16 Mtiles, 4 Ntiles, 32 K-slices), 1 wave32 per block
// A-frag 16x4 f32: lane L holds A[L%16][2*(L/16)+{0,1}]
// B-frag 4x16 f32: lane L holds B[2*(L/16)+{0,1}][L%16]
// C 16x16 f32:     lane L, vgpr v -> M = v + 8*(L/16), N = L%16
// ---------------------------------------------------------------
__global__ __launch_bounds__(32) void k_gemm(
    const float* __restrict__ fm, const float* __restrict__ pP,
    const float* __restrict__ pL,
    float* __restrict__ Amat, float* __restrict__ Bmat)
{
  const int lane  = threadIdx.x;
  const int mtile = blockIdx.x;     // 0..15
  const int ntile = blockIdx.y;     // 0..3
  const int kch   = blockIdx.z;     // 0..31  -> 128 rows of K each
  const int lo = lane & 15;
  const int h  = lane >> 4;
  const int mi = mtile * 16 + lo;   // global row of A (feature dim)
  const int ni = ntile * 16 + lo;   // global col (target-channel dim)

  v8f cp = {};
  v8f cl = {};
  const int kbase = kch * 128;
  for (int ks = 0; ks < 32; ++ks) {
    const int kk = kbase + ks * 4 + 2 * h;   // this lane's K pair base
    v2f av = { fm[(kk + 0) * FD + mi], fm[(kk + 1) * FD + mi] };
    v2f bp = { pP[(kk + 0) * TC + ni], pP[(kk + 1) * TC + ni] };
    v2f bl = { pL[(kk + 0) * TC + ni], pL[(kk + 1) * TC + ni] };
    cp = __builtin_amdgcn_wmma_f32_16x16x4_f32(false, av, false, bp,
                                               (short)0, cp, false, false);
    cl = __builtin_amdgcn_wmma_f32_16x16x4_f32(false, av, false, bl,
                                               (short)0, cl, false, false);
  }
#pragma unroll
  for (int v = 0; v < 8; ++v) {
    const int Mr = mtile * 16 + v + 8 * h;
    atomicAdd(&Amat[Mr * TC + ntile * 16 + lo], cp[v]);
    atomicAdd(&Bmat[Mr * TC + ntile * 16 + lo], cl[v]);
  }
}

// ---------------------------------------------------------------
// Kernel 3: gc = sum(A o B) over FD*TC = 16384 elements
// ---------------------------------------------------------------
__global__ __launch_bounds__(256) void k_gc(
    const float* __restrict__ Amat, const float* __restrict__ Bmat,
    float* __restrict__ gc)
{
  __shared__ float sh[256];
  const int idx = blockIdx.x * 256 + threadIdx.x;
  sh[threadIdx.x] = Amat[idx] * Bmat[idx];
  __syncthreads();
  for (int s = 128; s > 0; s >>= 1) {
    if (threadIdx.x < s) sh[threadIdx.x] += sh[threadIdx.x + s];
    __syncthreads();
  }
  if (threadIdx.x == 0) atomicAdd(gc, sh[0]);
}

// ---------------------------------------------------------------
// Kernel 4: tightness segment reduction (one block per code)
// ---------------------------------------------------------------
__global__ __launch_bounds__(FD) void k_seg(
    const float* __restrict__ segs, const float* __restrict__ segcnt,
    float* __restrict__ acc)
{
  __shared__ float sh[FD];
  const int g = blockIdx.x;
  const float v = segs[g * FD + threadIdx.x];
  sh[threadIdx.x] = v * v;
  __syncthreads();
  for (int s = FD / 2; s > 0; s >>= 1) {
    if (threadIdx.x < s) sh[threadIdx.x] += sh[threadIdx.x + s];
    __syncthreads();
  }
  if (threadIdx.x == 0) {
    const float s2  = sh[0];
    const float cnt = segcnt[g];
    const float mx  = fmaxf(cnt, 1.0f);
    atomicAdd(&acc[A_T2], s2 / mx);
    atomicAdd(&acc[A_T3], cnt * s2 / (mx * mx));
  }
}

// ---------------------------------------------------------------
// Kernel 5: finalize (serial; trivial work)
// ---------------------------------------------------------------
__global__ void k_final(const float* __restrict__ acc, float* __restrict__ out)
{
  if (threadIdx.x != 0 || blockIdx.x != 0) return;
  const float sm  = acc[A_SM];
  const float sa  = acc[A_SA];
  const float se  = acc[A_SE];
  const float sae = acc[A_SAE];
  const float gc  = acc[A_GC];
  const float t2  = acc[A_T2];
  const float t3  = acc[A_T3];

  float dvf = 0.0f;
  for (int i = 0; i < FD; ++i) dvf += acc[A_VF + i] * acc[A_VEF + i];
  float d1 = 0.0f, d2 = 0.0f;
  for (int c = 0; c < TC; ++c) {
    d1 += acc[A_VAP + c] * acc[A_VL + c];
    d2 += acc[A_VP + c]  * acc[A_VAL + c];
  }
  const float Df   = (float)DDIM;
  const float T_ae = sm * sae + sa * se;
  const float T_aC = (d1 + d2) / Df;
  const float T_Ge = 2.0f * dvf;
  const float T_GC = (2.0f / Df) * gc;
  const float S    = T_ae - T_aC - T_Ge + T_GC;

  const float diversity = -S / (sm * (sm - 1.0f));
  const float tightness = (sa - 2.0f * t2 + t3) / sm;
  const float total     = LAMB * diversity + LAMB * tightness;
  out[0] = total;
  out[1] = diversity;
  out[2] = tightness;
}

// ---------------------------------------------------------------
extern "C" void kernel_launch(void* const* d_in, const int* in_sizes, int n_in,
                              void* d_out, int out_size, void* d_ws, size_t ws_size,
                              hipStream_t stream) {
  (void)in_sizes; (void)n_in; (void)out_size; (void)ws_size;
  const float* features = (const float*)d_in[0];
  const float* targets  = (const float*)d_in[1];
  const unsigned char* mask = (const unsigned char*)d_in[2];

  float* ws     = (float*)d_ws;
  float* fm     = ws + OFF_FM;
  float* pP     = ws + OFF_P;
  float* pL     = ws + OFF_L;
  float* segs   = ws + OFF_SEGS;
  float* segcnt = ws + OFF_CNT;
  float* Amat   = ws + OFF_AM;
  float* Bmat   = ws + OFF_BM;
  float* acc    = ws + OFF_ACC;

  // zero all accumulator regions (segs..acc end)
  hipMemsetAsync(ws + OFF_SEGS, 0, (size_t)(WS_END - OFF_SEGS) * sizeof(float),
                 stream);

  k_prep<<<B, FD, 0, stream>>>(features, targets, mask, fm, pP, pL,
                               segs, segcnt, acc);
  k_gemm<<<dim3(FD / 16, TC / 16, 32), 32, 0, stream>>>(fm, pP, pL, Amat, Bmat);
  k_gc<<<(FD * TC) / 256, 256, 0, stream>>>(Amat, Bmat, acc + A_GC);
  k_seg<<<NSEG, FD, 0, stream>>>(segs, segcnt, acc);
  k_final<<<1, 1, 0, stream>>>(acc, (float*)d_out);
}